// MHA_78443282695044
// MI455X (gfx1250) — compile-verified
//
#include <hip/hip_runtime.h>

typedef __attribute__((ext_vector_type(2))) float v2f;
typedef __attribute__((ext_vector_type(8))) float v8f;
typedef __attribute__((ext_vector_type(4))) int   v4i;

// ---- CDNA5 async global->LDS path (ASYNCcnt), with safe fallback ----------
#if __has_builtin(__builtin_amdgcn_global_load_async_to_lds_b128)
typedef __attribute__((address_space(1))) v4i* asy_gptr;
typedef __attribute__((address_space(3))) v4i* asy_lptr;
#define ASYNC_COPY(g, l)                                                     \
    __builtin_amdgcn_global_load_async_to_lds_b128(                          \
        (asy_gptr)(void*)(g), (asy_lptr)(void*)(l), 0, 0)
#if __has_builtin(__builtin_amdgcn_s_wait_asynccnt)
#define ASYNC_WAIT(n) __builtin_amdgcn_s_wait_asynccnt(n)
#else
#define ASYNC_WAIT(n) asm volatile("s_wait_asynccnt %0" ::"i"(n) : "memory")
#endif
#else
#define ASYNC_COPY(g, l) (*(float4*)(void*)(l) = *(const float4*)(const void*)(g))
#define ASYNC_WAIT(n) do { } while (0)
#endif

static __device__ __forceinline__ v8f wmma4(v2f a, v2f b, v8f c) {
    // D = A(16x4,f32) * B(4x16,f32) + C(16x16,f32)
    return __builtin_amdgcn_wmma_f32_16x16x4_f32(
        /*neg_a=*/false, a, /*neg_b=*/false, b,
        /*c_mod=*/(short)0, c, /*reuse_a=*/false, /*reuse_b=*/false);
}

// ---------------------------------------------------------------------------
// Encoder GEMM (sync path: leaky_relu must be applied in flight):
//   h = leaky_relu(x) @ W_enc + b_enc     M=32768, Kd=256, N=512
// Block: 256 threads / 8 waves; C tile 64x64; K-chunks of 64 in LDS.
// ---------------------------------------------------------------------------
__global__ __launch_bounds__(256) void gemm_enc_f32(
    const float* __restrict__ A, const float* __restrict__ W,
    const float* __restrict__ bias, float* __restrict__ C,
    int Kd, int N)
{
    __shared__ float As[64 * 64];   // [m][k]
    __shared__ float Wt[64 * 64];   // [n][k] (transposed)

    const int tid = threadIdx.x;
    const int w  = tid >> 5;
    const int l  = tid & 31;
    const int lm = l & 15;
    const int g  = l >> 4;
    const int mb = blockIdx.x * 64;
    const int nb = blockIdx.y * 64;

    v8f acc[2] = {};

    const int nkc = Kd >> 6;
    for (int kc = 0; kc < nkc; ++kc) {
        #pragma unroll
        for (int i = 0; i < 4; ++i) {
            int f  = tid + i * 256;
            int m  = f >> 4;
            int k4 = (f & 15) << 2;
            float4 v = *(const float4*)&A[(long)(mb + m) * Kd + kc * 64 + k4];
            v.x = v.x >= 0.f ? v.x : 0.01f * v.x;
            v.y = v.y >= 0.f ? v.y : 0.01f * v.y;
            v.z = v.z >= 0.f ? v.z : 0.01f * v.z;
            v.w = v.w >= 0.f ? v.w : 0.01f * v.w;
            *(float4*)&As[m * 64 + k4] = v;
        }
        #pragma unroll
        for (int i = 0; i < 4; ++i) {
            int f  = tid + i * 256;
            int kl = f >> 4;
            int n4 = (f & 15) << 2;
            float4 v = *(const float4*)&W[(long)(kc * 64 + kl) * N + nb + n4];
            Wt[(n4 + 0) * 64 + kl] = v.x;
            Wt[(n4 + 1) * 64 + kl] = v.y;
            Wt[(n4 + 2) * 64 + kl] = v.z;
            Wt[(n4 + 3) * 64 + kl] = v.w;
        }
        __syncthreads();
        #pragma unroll
        for (int t = 0; t < 2; ++t) {
            int tile = w + t * 8;
            int mt = tile >> 2, nt = tile & 3;
            const float* ap = &As[(mt * 16 + lm) * 64];
            const float* bp = &Wt[(nt * 16 + lm) * 64];
            v8f c = acc[t];
            #pragma unroll
            for (int kk = 0; kk < 16; ++kk) {
                int k = kk * 4 + g * 2;
                c = wmma4(*(const v2f*)(ap + k), *(const v2f*)(bp + k), c);
            }
            acc[t] = c;
        }
        __syncthreads();
    }
    #pragma unroll
    for (int t = 0; t < 2; ++t) {
        int tile = w + t * 8;
        int mt = tile >> 2, nt = tile & 3;
        int col = nb + nt * 16 + lm;
        float badd = bias[col];
        #pragma unroll
        for (int r = 0; r < 8; ++r)
            C[(long)(mb + mt * 16 + r + 8 * g) * N + col] = acc[t][r] + badd;
    }
}

// ---------------------------------------------------------------------------
// Projection GEMM (Q/K/V): C = Amap @ W, Kd = N = 512, async double-buffered.
//   arow = ((row>>rshift)<<8)|(row&mask): rshift=8 identity, rshift=6 agents.
// LDS (dynamic, 64 KB): As[2][64*64] + Ws[2][64*64]; W staged untransposed,
// B fragment = two stride-64 b32 LDS reads (-> ds_load_2addr_stride64_b32).
// Per wave per buffer: 8 async b128 ops; wait keeps next buffer in flight.
// ---------------------------------------------------------------------------
__global__ __launch_bounds__(256) void gemm_proj_async(
    const float* __restrict__ A, const float* __restrict__ W,
    float* __restrict__ C, int rshift)
{
    extern __shared__ float gsm[];
    float* As0 = gsm;                 // 4096
    float* As1 = gsm + 4096;          // 4096
    float* Ws0 = gsm + 8192;          // 4096
    float* Ws1 = gsm + 12288;         // 4096

    const int tid = threadIdx.x;
    const int w  = tid >> 5;
    const int l  = tid & 31;
    const int lm = l & 15;
    const int g  = l >> 4;
    const int mb = blockIdx.x * 64;
    const int nb = blockIdx.y * 64;
    const int rmask = (1 << rshift) - 1;

    auto issue = [&](int kc, float* Asb, float* Wsb) {
        #pragma unroll
        for (int i = 0; i < 4; ++i) {
            int f = i * 256 + tid;                 // 2 rows / wave-instr
            int m = f >> 4, k4 = (f & 15) << 2;
            int row = mb + m;
            int arow = ((row >> rshift) << 8) | (row & rmask);
            ASYNC_COPY(&A[(long)arow * 512 + kc * 64 + k4], &Asb[m * 64 + k4]);
        }
        #pragma unroll
        for (int i = 0; i < 4; ++i) {
            int f = i * 256 + tid;
            int kl = f >> 4, n4 = (f & 15) << 2;
            ASYNC_COPY(&W[(long)(kc * 64 + kl) * 512 + nb + n4], &Wsb[kl * 64 + n4]);
        }
    };

    v8f acc[2] = {};
    issue(0, As0, Ws0);
    for (int kc = 0; kc < 8; ++kc) {
        float* Asc = (kc & 1) ? As1 : As0;
        float* Wsc = (kc & 1) ? Ws1 : Ws0;
        if (kc < 7) {
            issue(kc + 1, (kc & 1) ? As0 : As1, (kc & 1) ? Ws0 : Ws1);
            ASYNC_WAIT(8);            // drain current buffer, next stays in flight
        } else {
            ASYNC_WAIT(0);
        }
        __syncthreads();
        #pragma unroll
        for (int t = 0; t < 2; ++t) {
            int tile = w + t * 8;
            int mt = tile >> 2, nt = tile & 3;
            const float* ap = &Asc[(mt * 16 + lm) * 64];
            const float* bp = &Wsc[nt * 16 + lm];
            v8f c = acc[t];
            #pragma unroll
            for (int kk = 0; kk < 16; ++kk) {
                int k = kk * 4 + g * 2;
                v2f a = *(const v2f*)(ap + k);
                v2f b; b.x = bp[(k + 0) * 64]; b.y = bp[(k + 1) * 64];
                c = wmma4(a, b, c);
            }
            acc[t] = c;
        }
        __syncthreads();              // readers done before next issue reuses buf
    }
    #pragma unroll
    for (int t = 0; t < 2; ++t) {
        int tile = w + t * 8;
        int mt = tile >> 2, nt = tile & 3;
        int col = nb + nt * 16 + lm;
        #pragma unroll
        for (int r = 0; r < 8; ++r)
            C[(long)(mb + mt * 16 + r + 8 * g) * 512 + col] = acc[t][r];
    }
}

// ---------------------------------------------------------------------------
// Per-(head,batch) attention, async double-buffered streaming.
//   S = QK^T*scale -> softmax -> renorm(p/(sum+1e-12)) -> z = S@V, out += z/8.
// Dynamic LDS (229 KB): S 64x256 | buf0/buf1 each Qs(64x64)+Ks(256x64);
// phase B reuses buf0/buf1 for V chunks (256x64).
// ---------------------------------------------------------------------------
__global__ __launch_bounds__(256) void attn_wmma_f32(
    const float* __restrict__ Q, const float* __restrict__ K,
    const float* __restrict__ V, float* __restrict__ out, int accum)
{
    extern __shared__ float smem[];
    float* S    = smem;               // 16384
    float* buf0 = smem + 16384;       // 20480 = Qs(4096)+Ks(16384)
    float* buf1 = smem + 36864;       // 20480

    const int b   = blockIdx.x;
    const int tid = threadIdx.x;
    const int w  = tid >> 5;
    const int l  = tid & 31;
    const int lm = l & 15;
    const int g  = l >> 4;

    auto issueQK = [&](int dc, float* buf) {
        float* Qs = buf; float* Ks = buf + 4096;
        #pragma unroll
        for (int i = 0; i < 4; ++i) {     // Qs 64x64: 4 ops/wave
            int f = i * 256 + tid; int m = f >> 4, k4 = (f & 15) << 2;
            ASYNC_COPY(&Q[(long)(b * 64 + m) * 512 + dc * 64 + k4], &Qs[m * 64 + k4]);
        }
        #pragma unroll
        for (int i = 0; i < 16; ++i) {    // Ks 256x64: 16 ops/wave
            int f = i * 256 + tid; int e = f >> 4, k4 = (f & 15) << 2;
            ASYNC_COPY(&K[(long)(b * 256 + e) * 512 + dc * 64 + k4], &Ks[e * 64 + k4]);
        }
    };

    // ---------------- Phase A: S = Q K^T (reduce d=512, chunks of 64) ------
    v8f acc[8] = {};
    issueQK(0, buf0);
    for (int dc = 0; dc < 8; ++dc) {
        float* cur = (dc & 1) ? buf1 : buf0;
        if (dc < 7) {
            issueQK(dc + 1, (dc & 1) ? buf0 : buf1);
            ASYNC_WAIT(20);
        } else {
            ASYNC_WAIT(0);
        }
        __syncthreads();
        const float* Qs = cur;
        const float* Ks = cur + 4096;
        #pragma unroll
        for (int t = 0; t < 8; ++t) {     // 64 S-tiles / 8 waves
            int tile = w + t * 8;
            int mt = tile >> 4, et = tile & 15;
            const float* ap = &Qs[(mt * 16 + lm) * 64];
            const float* bp = &Ks[(et * 16 + lm) * 64];
            v8f c = acc[t];
            #pragma unroll
            for (int kk = 0; kk < 16; ++kk) {
                int k = kk * 4 + g * 2;
                c = wmma4(*(const v2f*)(ap + k), *(const v2f*)(bp + k), c);
            }
            acc[t] = c;
        }
        __syncthreads();
    }
    const float scale = 0.044194173824159216f;   // 1/sqrt(512)
    #pragma unroll
    for (int t = 0; t < 8; ++t) {
        int tile = w + t * 8;
        int mt = tile >> 4, et = tile & 15;
        #pragma unroll
        for (int r = 0; r < 8; ++r)
            S[(mt * 16 + r + 8 * g) * 256 + et * 16 + lm] = acc[t][r] * scale;
    }
    __syncthreads();

    // ---------------- softmax over e + reference's renorm ------------------
    if (tid < 64) {
        float* row = &S[tid * 256];
        float mx = row[0];
        for (int e = 1; e < 256; ++e) mx = fmaxf(mx, row[e]);
        float s = 0.f;
        for (int e = 0; e < 256; ++e) { float p = __expf(row[e] - mx); row[e] = p; s += p; }
        float inv = 1.f / s;
        float ps = 0.f;
        for (int e = 0; e < 256; ++e) { float p = row[e] * inv; row[e] = p; ps += p; }
        float inv2 = 1.f / (ps + 1e-12f);
        for (int e = 0; e < 256; ++e) row[e] *= inv2;
    }
    __syncthreads();

    // ---------------- Phase B: z = prob @ V (reduce e=256) -----------------
    auto issueV = [&](int dc, float* buf) {
        #pragma unroll
        for (int i = 0; i < 16; ++i) {    // Vs 256x64: 16 ops/wave
            int f = i * 256 + tid; int e = f >> 4, k4 = (f & 15) << 2;
            ASYNC_COPY(&V[(long)(b * 256 + e) * 512 + dc * 64 + k4], &buf[e * 64 + k4]);
        }
    };
    issueV(0, buf0);
    for (int dc = 0; dc < 8; ++dc) {
        const float* Vs = (dc & 1) ? buf1 : buf0;
        if (dc < 7) {
            issueV(dc + 1, (dc & 1) ? buf0 : buf1);
            ASYNC_WAIT(16);
        } else {
            ASYNC_WAIT(0);
        }
        __syncthreads();
        #pragma unroll
        for (int t = 0; t < 2; ++t) {     // 16 z-tiles / 8 waves
            int tile = w + t * 8;
            int mt = tile >> 2, nt = tile & 3;
            const float* ap = &S[(mt * 16 + lm) * 256];
            v8f c = {};
            #pragma unroll 8
            for (int kk = 0; kk < 64; ++kk) {
                int k = kk * 4 + g * 2;
                v2f a = *(const v2f*)(ap + k);
                v2f bb;
                bb.x = Vs[(k + 0) * 64 + nt * 16 + lm];
                bb.y = Vs[(k + 1) * 64 + nt * 16 + lm];
                c = wmma4(a, bb, c);
            }
            int col = dc * 64 + nt * 16 + lm;
            #pragma unroll
            for (int r = 0; r < 8; ++r) {
                long idx = (long)(b * 64 + mt * 16 + r + 8 * g) * 512 + col;
                float z = c[r] * 0.125f;       // mean over 8 heads
                if (accum) out[idx] += z; else out[idx] = z;
            }
        }
        __syncthreads();
    }
}

extern "C" void kernel_launch(void* const* d_in, const int* in_sizes, int n_in,
                              void* d_out, int out_size, void* d_ws, size_t ws_size,
                              hipStream_t stream)
{
    (void)in_sizes; (void)n_in; (void)out_size; (void)ws_size;
    const float* x    = (const float*)d_in[0];   // (128,256,256)
    const float* Wenc = (const float*)d_in[1];   // (256,512)
    const float* benc = (const float*)d_in[2];   // (512)
    const float* WQ   = (const float*)d_in[3];   // (8,512,512)
    const float* WK   = (const float*)d_in[4];
    const float* WV   = (const float*)d_in[5];
    float* out = (float*)d_out;                  // (128,64,512)

    float* h  = (float*)d_ws;                    // 32768*512
    float* Qw = h  + (size_t)32768 * 512;        //  8192*512
    float* Kw = Qw + (size_t)8192  * 512;        // 32768*512
    float* Vw = Kw + (size_t)32768 * 512;        // 32768*512

    const dim3 blk(256);
    const size_t proj_lds = 16384 * sizeof(float);                            // 64 KB
    const size_t attn_lds = (16384 + 2 * 20480) * sizeof(float);              // 229 KB

    // Encoder: h = leaky_relu(x) @ W_enc + b_enc   (M=32768,K=256,N=512)
    gemm_enc_f32<<<dim3(512, 8), blk, 0, stream>>>(x, Wenc, benc, h, 256, 512);

    for (int n = 0; n < 8; ++n) {
        const float* wq = WQ + (size_t)n * 512 * 512;
        const float* wk = WK + (size_t)n * 512 * 512;
        const float* wv = WV + (size_t)n * 512 * 512;
        gemm_proj_async<<<dim3(128, 8), blk, proj_lds, stream>>>(h, wq, Qw, 6);
        gemm_proj_async<<<dim3(512, 8), blk, proj_lds, stream>>>(h, wk, Kw, 8);
        gemm_proj_async<<<dim3(512, 8), blk, proj_lds, stream>>>(h, wv, Vw, 8);
        attn_wmma_f32<<<dim3(128), blk, attn_lds, stream>>>(Qw, Kw, Vw, out, n > 0 ? 1 : 0);
    }
}